// OFDMProcessModule_53644141527076
// MI455X (gfx1250) — compile-verified
//
#include <hip/hip_runtime.h>

typedef float v2f __attribute__((ext_vector_type(2)));
typedef float v4f __attribute__((ext_vector_type(4)));
typedef float v8f __attribute__((ext_vector_type(8)));

#define FFT            64
#define ROWS_PER_BLOCK 128
#define THREADS        256
#define BSTRIDE        68    // padded stride for B matrices (bank-conflict free)
#define DSTRIDE        132   // padded stride for staged input rows
#define FSTRIDE        66    // padded stride for staged fr/fi rows

#define OFF_BC   0
#define OFF_BS   (64 * BSTRIDE)
#define OFF_DATA (2 * 64 * BSTRIDE)
#define LDS_FLOATS (OFF_DATA + ROWS_PER_BLOCK * DSTRIDE)   // 25600 floats = 100 KB

__global__ __launch_bounds__(THREADS)
void ofdm_eq_kernel(const float* __restrict__ x, float* __restrict__ out)
{
    extern __shared__ float smem[];
    float* Bc   = smem + OFF_BC;    // Bc[n*BSTRIDE + k] =  cos(2*pi*n*k/64)  (symmetric)
    float* Bs   = smem + OFF_BS;    // Bs[n*BSTRIDE + k] = -sin(2*pi*n*k/64)  (symmetric)
    float* data = smem + OFF_DATA;  // staged input rows, later reused for fr/fi staging

    const int t    = threadIdx.x;
    const int lane = t & 31;
    const int wave = t >> 5;
    const int half = (lane >> 4) & 1;   // wave32: low/high half selects K sub-pair
    const int lm   = lane & 15;

    // ---- Phase 0a: build DFT matrices in LDS (period-64 exact index reduction) ----
    #pragma unroll
    for (int i = 0; i < 16; ++i) {
        int idx = t + i * THREADS;           // 0..4095
        int n = idx >> 6, k = idx & 63;
        int m = (n * k) & 63;                // exact mod-64 phase
        float ang = (float)m * 0.09817477042468103f;  // 2*pi/64
        float s, c;
        __sincosf(ang, &s, &c);
        Bc[n * BSTRIDE + k] = c;
        Bs[n * BSTRIDE + k] = -s;
    }

    // ---- Phase 0b: coalesced load of 128 rows x 128 floats into LDS ----
    const float* xblk = x + (size_t)blockIdx.x * (ROWS_PER_BLOCK * 2 * FFT);
    #pragma unroll
    for (int i = 0; i < 16; ++i) {
        int f4 = t + i * THREADS;            // float4 index 0..4095
        int r  = f4 >> 5;                    // row (32 float4 per row)
        int c4 = f4 & 31;
        v4f vec = *(const v4f*)(xblk + (size_t)f4 * 4);
        *(v4f*)(data + r * DSTRIDE + c4 * 4) = vec;
    }
    __syncthreads();

    // ---- Phase 1: load A fragments (16 rows x 64 K) into registers ----
    // A layout (f32 16x4): VGPR0 = K {0|2}, VGPR1 = K {1|3}; lanes 0-15 K-low, 16-31 K-high.
    const int rb = wave * 16;                // block-local row base of this wave's tile
    v2f ar[16], ai[16], an[16];
    {
        const float* rowp = data + (rb + lm) * DSTRIDE;
        #pragma unroll
        for (int k = 0; k < 16; ++k) {
            int col = 4 * k + 2 * half;
            ar[k] = *(const v2f*)(rowp + col);        // xr
            ai[k] = *(const v2f*)(rowp + 64 + col);   // xi
            an[k] = -ai[k];                           // -xi (f32 WMMA has no A-neg modifier)
        }
    }
    __syncthreads();   // all waves done reading staged input before fr/fi overwrite it

    float* frs = data;                                  // [128][FSTRIDE]
    float* fis = data + ROWS_PER_BLOCK * FSTRIDE;       // [128][FSTRIDE]

    // ---- Phase 2: DFT via fp32 WMMA. fr = xr*Bc + (-xi)*Bs ; fi = xi*Bc + xr*Bs ----
    #pragma unroll
    for (int n = 0; n < 4; ++n) {
        v8f accr = {};
        v8f acci = {};
        const float* bcp = Bc + (n * 16 + lm) * BSTRIDE + 2 * half;
        const float* bsp = Bs + (n * 16 + lm) * BSTRIDE + 2 * half;
        #pragma unroll
        for (int k = 0; k < 16; ++k) {
            v2f bc = *(const v2f*)(bcp + 4 * k);
            v2f bs = *(const v2f*)(bsp + 4 * k);
            accr = __builtin_amdgcn_wmma_f32_16x16x4_f32(false, ar[k], false, bc,
                                                         (short)0, accr, false, false);
            accr = __builtin_amdgcn_wmma_f32_16x16x4_f32(false, an[k], false, bs,
                                                         (short)0, accr, false, false);
            acci = __builtin_amdgcn_wmma_f32_16x16x4_f32(false, ai[k], false, bc,
                                                         (short)0, acci, false, false);
            acci = __builtin_amdgcn_wmma_f32_16x16x4_f32(false, ar[k], false, bs,
                                                         (short)0, acci, false, false);
        }
        // Stage C/D tile: VGPR v holds (M=v, lanes 0-15) and (M=v+8, lanes 16-31), N=lane%16.
        #pragma unroll
        for (int v = 0; v < 8; ++v) {
            int r = rb + v + 8 * half;
            frs[r * FSTRIDE + n * 16 + lm] = accr[v];
            fis[r * FSTRIDE + n * 16 + lm] = acci[v];
        }
    }
    __syncthreads();

    // ---- Phase 3: pilot interpolation + equalization, coalesced fp32 output ----
    float* outblk = out + (size_t)blockIdx.x * (ROWS_PER_BLOCK * 2 * FFT);
    #pragma unroll 4
    for (int i = 0; i < 32; ++i) {
        int e = t + i * THREADS;             // 0..8191
        int r = e >> 6, c = e & 63;
        const float* fr_row = frs + r * FSTRIDE;
        const float* fi_row = fis + r * FSTRIDE;
        float fr = fr_row[c];
        float fi = fi_row[c];

        float hr, hi;
        if (c <= 11)      { hr = fr_row[11]; hi = fi_row[11]; }
        else if (c >= 53) { hr = fr_row[53]; hi = fi_row[53]; }
        else {
            int seg  = (c - 11) / 14;        // 0..2
            int base = 11 + seg * 14;
            float a  = (float)(c - base) * (1.0f / 14.0f);
            float b  = 1.0f - a;
            hr = b * fr_row[base] + a * fr_row[base + 14];
            hi = b * fi_row[base] + a * fi_row[base + 14];
        }
        float denom = hr * hr + hi * hi + 1e-8f;
        float er = (fr * hr + fi * hi) / denom;
        float ei = (fi * hr - fr * hi) / denom;
        outblk[r * 128 + c]      = er;
        outblk[r * 128 + 64 + c] = ei;
    }
}

extern "C" void kernel_launch(void* const* d_in, const int* in_sizes, int n_in,
                              void* d_out, int out_size, void* d_ws, size_t ws_size,
                              hipStream_t stream)
{
    (void)n_in; (void)d_ws; (void)ws_size; (void)out_size;
    const float* x = (const float*)d_in[0];
    float* out = (float*)d_out;

    int total_f = in_sizes[0];               // N * 2 * 64
    int rows    = total_f / (2 * FFT);       // N = 262144
    int blocks  = rows / ROWS_PER_BLOCK;     // 2048

    size_t shmem = (size_t)LDS_FLOATS * sizeof(float);   // 100 KB dynamic LDS
    static bool attr_set = false;
    if (!attr_set) {
        // Allow >64KB dynamic LDS where required; harmless if it fails.
        (void)hipFuncSetAttribute((const void*)ofdm_eq_kernel,
                                  hipFuncAttributeMaxDynamicSharedMemorySize,
                                  (int)shmem);
        attr_set = true;
    }
    ofdm_eq_kernel<<<blocks, THREADS, shmem, stream>>>(x, out);
}